// GNN_foraging_extended_71536975282841
// MI455X (gfx1250) — compile-verified
//
#include <hip/hip_runtime.h>

typedef __attribute__((ext_vector_type(2))) float v2f;
typedef __attribute__((ext_vector_type(8))) float v8f;

#define N_NODES 100000
#define N_EDGES 1000000
#define HDIM    64
#define DIN     32
#define DOUT    8
#define BN_EPS  1e-5f

// ---------------------------------------------------------------- utilities
__global__ void fill_f32(float* __restrict__ p, float v, int n) {
    int i = blockIdx.x * blockDim.x + threadIdx.x;
    if (i < n) p[i] = v;
}

__global__ void count_deg(const long long* __restrict__ dst, float* __restrict__ deg, int e) {
    int i = blockIdx.x * blockDim.x + threadIdx.x;
    if (i < e) atomicAdd(&deg[(int)dst[i]], 1.0f);
}

__global__ void finish_dinv(float* __restrict__ d, int n) {
    int i = blockIdx.x * blockDim.x + threadIdx.x;
    if (i < n) d[i] = rsqrtf(d[i]);
}

// ------------------------------------------------- dense transform: m = A @ W
// A: [N, FIN] row-major, W: [FIN, 64] row-major, out: [N, 64].
// One block = 4 waves; each wave produces one 16x16 tile via V_WMMA_F32_16X16X4_F32.
template <int FIN>
__global__ __launch_bounds__(128) void gemm_wmma(const float* __restrict__ A,
                                                 const float* __restrict__ W,
                                                 float* __restrict__ out) {
    const int wave = threadIdx.x >> 5;
    const int lane = threadIdx.x & 31;
    const int row0 = blockIdx.x << 4;   // 16-row tile (N % 16 == 0)
    const int col0 = wave << 4;         // 4 tiles cover 64 cols
    const int mn   = lane & 15;         // M for A-lanes, N for B-lanes
    const int hi   = lane >> 4;         // 0 or 1
    const int koff = hi << 1;           // K sub-offset: 0 or 2

    const float* __restrict__ arow = A + (size_t)(row0 + mn) * FIN;
    v8f c = {};
#pragma unroll
    for (int kb = 0; kb < FIN; kb += 4) {
        v2f a, b;
        a.x = arow[kb + koff];
        a.y = arow[kb + koff + 1];
        b.x = W[(size_t)(kb + koff) * HDIM + col0 + mn];
        b.y = W[(size_t)(kb + koff + 1) * HDIM + col0 + mn];
        c = __builtin_amdgcn_wmma_f32_16x16x4_f32(false, a, false, b,
                                                  (short)0, c, false, false);
    }
    const int col   = col0 + mn;
    const int rbase = row0 + (hi << 3); // lanes 16..31 hold M = r + 8
#pragma unroll
    for (int r = 0; r < 8; ++r)
        out[(size_t)(rbase + r) * HDIM + col] = c[r];
}

// ------------------------------------------- edge scatter: agg[dst] += en*m[src]
// One thread per (edge, 4-feature chunk): float4 gather + 4 f32 atomics (L2-resident).
__global__ __launch_bounds__(256) void scatter_edges(const long long* __restrict__ src,
                                                     const long long* __restrict__ dst,
                                                     const float* __restrict__ dinv,
                                                     const float* __restrict__ m,
                                                     float* __restrict__ agg) {
    long long t = (long long)blockIdx.x * blockDim.x + threadIdx.x;
    if (t >= (long long)N_EDGES * 16) return;
    const int e  = (int)(t >> 4);
    const int f4 = (int)(t & 15) << 2;
    const int s = (int)src[e];
    const int d = (int)dst[e];
    const float en = dinv[s] * dinv[d];
    const float4 mv = *(const float4*)(m + (size_t)s * HDIM + f4);
    float* base = agg + (size_t)d * HDIM + f4;
    atomicAdd(base + 0, en * mv.x);
    atomicAdd(base + 1, en * mv.y);
    atomicAdd(base + 2, en * mv.z);
    atomicAdd(base + 3, en * mv.w);
}

// --------------------------------- h = relu(agg + self_norm*m + b), float4 wide
__global__ __launch_bounds__(256) void combine_relu(const float* __restrict__ agg,
                                                    const float* __restrict__ m,
                                                    const float* __restrict__ dinv,
                                                    const float* __restrict__ bias,
                                                    float* __restrict__ h) {
    int t = blockIdx.x * blockDim.x + threadIdx.x;
    if (t >= N_NODES * 16) return;
    const int i  = t >> 4;
    const int f4 = (t & 15) << 2;
    const float di = dinv[i];
    const float sn = di * di;
    const float4 a = *(const float4*)(agg + (size_t)i * HDIM + f4);
    const float4 mm = *(const float4*)(m + (size_t)i * HDIM + f4);
    const float4 b = *(const float4*)(bias + f4);
    float4 r;
    r.x = fmaxf(a.x + sn * mm.x + b.x, 0.0f);
    r.y = fmaxf(a.y + sn * mm.y + b.y, 0.0f);
    r.z = fmaxf(a.z + sn * mm.z + b.z, 0.0f);
    r.w = fmaxf(a.w + sn * mm.w + b.w, 0.0f);
    *(float4*)(h + (size_t)i * HDIM + f4) = r;
}

// ------------------- BN stats: stats[0..63]=sum, stats[64..127]=sumsq (atomics)
__global__ __launch_bounds__(64) void bn_stats(const float* __restrict__ h,
                                               float* __restrict__ stats) {
    const int f  = threadIdx.x;              // feature 0..63
    const int r0 = blockIdx.x * 256;
    const int r1 = (r0 + 256 < N_NODES) ? r0 + 256 : N_NODES;
    float s = 0.0f, sq = 0.0f;
    for (int i = r0; i < r1; ++i) {
        const float v = h[(size_t)i * HDIM + f];
        s += v; sq += v * v;
    }
    atomicAdd(&stats[f], s);
    atomicAdd(&stats[HDIM + f], sq);
}

__global__ __launch_bounds__(256) void bn_apply(float* __restrict__ h,
                                                const float* __restrict__ stats,
                                                const float* __restrict__ g,
                                                const float* __restrict__ be) {
    int t = blockIdx.x * blockDim.x + threadIdx.x;
    if (t >= N_NODES * 16) return;
    const int i  = t >> 4;
    const int f4 = (t & 15) << 2;
    const float inv_n = 1.0f / (float)N_NODES;
    float4 v = *(float4*)(h + (size_t)i * HDIM + f4);
    float* vp = (float*)&v;
#pragma unroll
    for (int c = 0; c < 4; ++c) {
        const int f = f4 + c;
        const float mu  = stats[f] * inv_n;
        const float var = stats[HDIM + f] * inv_n - mu * mu;
        const float sc  = g[f] * rsqrtf(var + BN_EPS);
        vp[c] = (vp[c] - mu) * sc + be[f];
    }
    *(float4*)(h + (size_t)i * HDIM + f4) = v;
}

// ------------------------- fc: out = h @ fcW + fcb  (64x8, WMMA with padded N)
__global__ __launch_bounds__(32) void fc_wmma(const float* __restrict__ h,
                                              const float* __restrict__ fcW,
                                              const float* __restrict__ fcb,
                                              float* __restrict__ out) {
    const int lane = threadIdx.x & 31;
    const int row0 = blockIdx.x << 4;
    const int mn   = lane & 15;
    const int hi   = lane >> 4;
    const int koff = hi << 1;
    // keep loads in-bounds & EXEC uniform: index with (mn&7), select with sel
    const float sel = (mn < DOUT) ? 1.0f : 0.0f;
    const int   nc  = mn & 7;
    const float* __restrict__ hrow = h + (size_t)(row0 + mn) * HDIM;
    v8f c = {};
#pragma unroll
    for (int kb = 0; kb < HDIM; kb += 4) {
        v2f a, b;
        a.x = hrow[kb + koff];
        a.y = hrow[kb + koff + 1];
        b.x = sel * fcW[(size_t)(kb + koff) * DOUT + nc];
        b.y = sel * fcW[(size_t)(kb + koff + 1) * DOUT + nc];
        c = __builtin_amdgcn_wmma_f32_16x16x4_f32(false, a, false, b,
                                                  (short)0, c, false, false);
    }
    if (mn < DOUT) {
        const float bb = fcb[mn];
        const int rbase = row0 + (hi << 3);
#pragma unroll
        for (int r = 0; r < 8; ++r)
            out[(size_t)(rbase + r) * DOUT + mn] = c[r] + bb;
    }
}

// ================================================================== host side
static inline int cdiv(long long a, long long b) { return (int)((a + b - 1) / b); }

extern "C" void kernel_launch(void* const* d_in, const int* in_sizes, int n_in,
                              void* d_out, int out_size, void* d_ws, size_t ws_size,
                              hipStream_t stream) {
    const float*     x    = (const float*)d_in[0];
    const long long* ei   = (const long long*)d_in[1];
    const long long* src  = ei;
    const long long* dstp = ei + N_EDGES;
    const float* W[3]  = {(const float*)d_in[2], (const float*)d_in[6],  (const float*)d_in[10]};
    const float* bb[3] = {(const float*)d_in[3], (const float*)d_in[7],  (const float*)d_in[11]};
    const float* gg[3] = {(const float*)d_in[4], (const float*)d_in[8],  (const float*)d_in[12]};
    const float* be[3] = {(const float*)d_in[5], (const float*)d_in[9],  (const float*)d_in[13]};
    const float* fcW = (const float*)d_in[14];
    const float* fcb = (const float*)d_in[15];
    float* out = (float*)d_out;

    // workspace carve-up (all 256B aligned)
    char* ws = (char*)d_ws;
    const size_t FEAT_BYTES = (size_t)N_NODES * HDIM * sizeof(float); // 25.6 MB
    float* dinv  = (float*)(ws);                                      // N floats (deg -> dinv)
    float* m     = (float*)(ws + (512ull << 10));
    float* agg   = (float*)(ws + (512ull << 10) + FEAT_BYTES);
    float* h     = (float*)(ws + (512ull << 10) + 2 * FEAT_BYTES);
    float* stats = (float*)(ws + (512ull << 10) + 3 * FEAT_BYTES);    // 128 floats
    (void)ws_size; (void)n_in; (void)in_sizes; (void)out_size;

    const int TILES   = N_NODES / 16;      // 6250 (exact)
    const int ELEM4   = N_NODES * 16;      // N*64/4 float4 work-items
    const int SCAT_T  = 256;

    // degree / normalization
    fill_f32<<<cdiv(N_NODES, 256), 256, 0, stream>>>(dinv, 1.0f, N_NODES);
    count_deg<<<cdiv(N_EDGES, 256), 256, 0, stream>>>(dstp, dinv, N_EDGES);
    finish_dinv<<<cdiv(N_NODES, 256), 256, 0, stream>>>(dinv, N_NODES);

    const float* layer_in = x;
    for (int l = 0; l < 3; ++l) {
        if (l == 0)
            gemm_wmma<DIN><<<TILES, 128, 0, stream>>>(layer_in, W[l], m);
        else
            gemm_wmma<HDIM><<<TILES, 128, 0, stream>>>(layer_in, W[l], m);

        fill_f32<<<cdiv((long long)N_NODES * HDIM, 256), 256, 0, stream>>>(agg, 0.0f, N_NODES * HDIM);
        fill_f32<<<1, 128, 0, stream>>>(stats, 0.0f, 2 * HDIM);

        scatter_edges<<<cdiv((long long)N_EDGES * 16, SCAT_T), SCAT_T, 0, stream>>>(src, dstp, dinv, m, agg);
        combine_relu<<<cdiv(ELEM4, 256), 256, 0, stream>>>(agg, m, dinv, bb[l], h);
        bn_stats<<<cdiv(N_NODES, 256), 64, 0, stream>>>(h, stats);
        bn_apply<<<cdiv(ELEM4, 256), 256, 0, stream>>>(h, stats, gg[l], be[l]);
        layer_in = h;
    }

    fc_wmma<<<TILES, 32, 0, stream>>>(h, fcW, fcb, out);
}